// GraphResNetBlock_88510686036732
// MI455X (gfx1250) — compile-verified
//
#include <hip/hip_runtime.h>

#define HEADS 4
#define CPH 32
#define OC 128
#define NEG_SLOPE 0.2f
#define EPS_BN 1e-5f

typedef __attribute__((ext_vector_type(2))) float v2f;
typedef __attribute__((ext_vector_type(8))) float v8f;

// monotone float <-> uint encoding: float segment-max via atomicMax(u32)
__device__ __forceinline__ unsigned ordF(float f) {
    unsigned u = __float_as_uint(f);
    return (u & 0x80000000u) ? ~u : (u | 0x80000000u);
}
__device__ __forceinline__ float unordF(unsigned u) {
    return __uint_as_float((u & 0x80000000u) ? (u & 0x7fffffffu) : ~u);
}
__device__ __forceinline__ float lrelu(float v) { return v >= 0.f ? v : NEG_SLOPE * v; }

// ---------------------------------------------------------------------------
// H[N,128] = X[N,128] @ W[128,128]  via V_WMMA_F32_16X16X4_F32.
// block = 256 thr = 8 waves; block computes 64 rows x 128 cols.
// Each wave owns one 16-col strip and 4 M-tiles sharing each B fragment.
// ---------------------------------------------------------------------------
__global__ __launch_bounds__(256) void gemm_wmma_kernel(
    const float* __restrict__ X, const float* __restrict__ W,
    float* __restrict__ H, int N) {
    __shared__ float As[64][128];                 // 32 KB of the 320 KB LDS
    const int m0 = blockIdx.x * 64;

    for (int t = threadIdx.x; t < 64 * 32; t += 256) {
        int r  = t >> 5;
        int c4 = (t & 31) << 2;
        int row = m0 + r;
        float4 v = make_float4(0.f, 0.f, 0.f, 0.f);
        if (row < N) v = *reinterpret_cast<const float4*>(X + (size_t)row * OC + c4);
        *reinterpret_cast<float4*>(&As[r][c4]) = v;
    }
    __syncthreads();

    const int wave = threadIdx.x >> 5;
    const int lane = threadIdx.x & 31;
    const int ln   = lane & 15;   // M index (A) / N index (B,C)
    const int kh   = lane >> 4;   // 0 -> K pair {0,1}, 1 -> K pair {2,3}
    const int n0   = wave * 16;

    v8f acc[4];
#pragma unroll
    for (int t = 0; t < 4; ++t) acc[t] = (v8f){};

#pragma unroll
    for (int k0 = 0; k0 < 128; k0 += 4) {
        v2f b;
        // B 4x16 f32 layout: lane = (k>=2?16:0)+n, vgpr = k%2
        b.x = W[(k0 + kh * 2 + 0) * OC + n0 + ln];
        b.y = W[(k0 + kh * 2 + 1) * OC + n0 + ln];
#pragma unroll
        for (int t = 0; t < 4; ++t) {
            v2f a;
            // A 16x4 f32 layout: lane = (k>=2?16:0)+m, vgpr = k%2
            a.x = As[t * 16 + ln][k0 + kh * 2 + 0];
            a.y = As[t * 16 + ln][k0 + kh * 2 + 1];
            acc[t] = __builtin_amdgcn_wmma_f32_16x16x4_f32(false, a, false, b,
                                                           (short)0, acc[t], false, false);
        }
    }
    // C/D 16x16 f32: VGPR g, lanes 0-15 -> M=g, lanes 16-31 -> M=8+g, N=lane%16
#pragma unroll
    for (int t = 0; t < 4; ++t) {
#pragma unroll
        for (int g = 0; g < 8; ++g) {
            int row = m0 + t * 16 + g + kh * 8;
            if (row < N) H[(size_t)row * OC + n0 + ln] = acc[t][g];
        }
    }
}

// ---------------------------------------------------------------------------
// alpha_s[n,h] = sum_c h[n,h,c]*a_src[h,c]  (one wave per node)
// ---------------------------------------------------------------------------
__global__ __launch_bounds__(256) void alphas_kernel(
    const float* __restrict__ H, const float* __restrict__ a_src,
    const float* __restrict__ a_dst, float* __restrict__ as_out,
    float* __restrict__ ad_out, int N) {
    int n    = (int)((blockIdx.x * (size_t)blockDim.x + threadIdx.x) >> 5);
    int lane = threadIdx.x & 31;
    if (n >= N) return;
    int cb   = lane * 4;
    int head = lane >> 3;
    int cc   = (lane & 7) * 4;
    float4 h = *reinterpret_cast<const float4*>(H + (size_t)n * OC + cb);
    float4 s = *reinterpret_cast<const float4*>(a_src + head * CPH + cc);
    float4 d = *reinterpret_cast<const float4*>(a_dst + head * CPH + cc);
    float ps = h.x * s.x + h.y * s.y + h.z * s.z + h.w * s.w;
    float pd = h.x * d.x + h.y * d.y + h.z * d.z + h.w * d.w;
#pragma unroll
    for (int o = 1; o < 8; o <<= 1) {
        ps += __shfl_xor(ps, o, 32);
        pd += __shfl_xor(pd, o, 32);
    }
    if ((lane & 7) == 0) {
        as_out[n * HEADS + head] = ps;
        ad_out[n * HEADS + head] = pd;
    }
}

// ---------------------------------------------------------------------------
// CSR build: degree count -> two-level scan -> rowstart[N+1]
// ---------------------------------------------------------------------------
__global__ void deg_count_kernel(const int* __restrict__ dst, int* __restrict__ deg,
                                 int E, int EN) {
    int i = blockIdx.x * blockDim.x + threadIdx.x;
    if (i >= EN) return;
    int d = (i < E) ? dst[i] : i - E;
    atomicAdd(&deg[d], 1);
}

__global__ __launch_bounds__(256) void scan_partial_kernel(
    const int* __restrict__ deg, int* __restrict__ partial, int N) {
    __shared__ int sh[256];
    int t = threadIdx.x;
    int i = blockIdx.x * 256 + t;
    sh[t] = (i < N) ? deg[i] : 0;
    __syncthreads();
#pragma unroll
    for (int o = 128; o > 0; o >>= 1) {
        if (t < o) sh[t] += sh[t + o];
        __syncthreads();
    }
    if (t == 0) partial[blockIdx.x] = sh[0];
}

// single block: exclusive-scan the per-block partials (chunks of 512)
__global__ __launch_bounds__(512) void scan_offsets_kernel(
    const int* __restrict__ partial, int* __restrict__ partOff, int nb,
    int* __restrict__ rowstart, int N, int EN) {
    __shared__ int sh[2][512];
    __shared__ int carry;
    int t = threadIdx.x;
    if (t == 0) carry = 0;
    __syncthreads();
    for (int base = 0; base < nb; base += 512) {
        int idx = base + t;
        int v = (idx < nb) ? partial[idx] : 0;
        int buf = 0;
        sh[buf][t] = v;
        __syncthreads();
        for (int o = 1; o < 512; o <<= 1) {
            int x = sh[buf][t];
            if (t >= o) x += sh[buf][t - o];
            sh[buf ^ 1][t] = x;
            buf ^= 1;
            __syncthreads();
        }
        int incl = sh[buf][t];
        if (idx < nb) partOff[idx] = carry + incl - v;   // exclusive
        __syncthreads();
        if (t == 0) carry += sh[buf][511];
        __syncthreads();
    }
    if (t == 0) rowstart[N] = EN;
}

__global__ __launch_bounds__(256) void scan_write_kernel(
    const int* __restrict__ deg, const int* __restrict__ partOff,
    int* __restrict__ rowstart, int N) {
    __shared__ int sh[2][256];
    int t = threadIdx.x;
    int i = blockIdx.x * 256 + t;
    int v = (i < N) ? deg[i] : 0;
    int buf = 0;
    sh[buf][t] = v;
    __syncthreads();
    for (int o = 1; o < 256; o <<= 1) {
        int x = sh[buf][t];
        if (t >= o) x += sh[buf][t - o];
        sh[buf ^ 1][t] = x;
        buf ^= 1;
        __syncthreads();
    }
    if (i < N) rowstart[i] = partOff[blockIdx.x] + sh[buf][t] - v;  // exclusive
}

// ---------------------------------------------------------------------------
// pass A: segment max of leaky(logits) per (dst, head)
// ---------------------------------------------------------------------------
__global__ void edge_max_kernel(const int* __restrict__ src, const int* __restrict__ dst,
                                const float* __restrict__ as_, const float* __restrict__ ad_,
                                unsigned* __restrict__ mmax, int E, int EN) {
    int i = blockIdx.x * blockDim.x + threadIdx.x;
    if (i >= EN) return;
    int s, d;
    if (i < E) { s = src[i]; d = dst[i]; } else { s = d = i - E; }
    float4 a = *reinterpret_cast<const float4*>(as_ + (size_t)s * HEADS);
    float4 b = *reinterpret_cast<const float4*>(ad_ + (size_t)d * HEADS);
    unsigned* mp = mmax + (size_t)d * HEADS;
    atomicMax(mp + 0, ordF(lrelu(a.x + b.x)));
    atomicMax(mp + 1, ordF(lrelu(a.y + b.y)));
    atomicMax(mp + 2, ordF(lrelu(a.z + b.z)));
    atomicMax(mp + 3, ordF(lrelu(a.w + b.w)));
}

// ---------------------------------------------------------------------------
// pass B: e = exp(logit - m[dst]); denom[dst] += e; scatter e + src into
// dst-sorted CSR slots (fused fill).
// ---------------------------------------------------------------------------
__global__ void edge_exp_fill_kernel(
    const int* __restrict__ src, const int* __restrict__ dst,
    const float* __restrict__ as_, const float* __restrict__ ad_,
    const unsigned* __restrict__ mmax, float* __restrict__ denom,
    int* __restrict__ cursor, float* __restrict__ esorted,
    int* __restrict__ esrc, int E, int EN) {
    int i = blockIdx.x * blockDim.x + threadIdx.x;
    if (i >= EN) return;
    int s, d;
    if (i < E) { s = src[i]; d = dst[i]; } else { s = d = i - E; }
    float4 a = *reinterpret_cast<const float4*>(as_ + (size_t)s * HEADS);
    float4 b = *reinterpret_cast<const float4*>(ad_ + (size_t)d * HEADS);
    const unsigned* mp = mmax + (size_t)d * HEADS;
    float4 e;
    e.x = __expf(lrelu(a.x + b.x) - unordF(mp[0]));
    e.y = __expf(lrelu(a.y + b.y) - unordF(mp[1]));
    e.z = __expf(lrelu(a.z + b.z) - unordF(mp[2]));
    e.w = __expf(lrelu(a.w + b.w) - unordF(mp[3]));
    float* dp = denom + (size_t)d * HEADS;
    atomicAdd(dp + 0, e.x);
    atomicAdd(dp + 1, e.y);
    atomicAdd(dp + 2, e.z);
    atomicAdd(dp + 3, e.w);
    int pos = atomicAdd(&cursor[d], 1);
    *reinterpret_cast<float4*>(esorted + (size_t)pos * HEADS) = e;
    esrc[pos] = s;
}

// ---------------------------------------------------------------------------
// heavy pass, gather style: one wave per dst node, 4 channels per lane,
// register accumulation, single coalesced store. Zero float atomics.
// ---------------------------------------------------------------------------
__global__ __launch_bounds__(256) void agg_csr_kernel(
    const int* __restrict__ rowstart, const int* __restrict__ esrc,
    const float* __restrict__ esorted, const float* __restrict__ denom,
    const float* __restrict__ H, float* __restrict__ agg, int N) {
    int n    = (int)((blockIdx.x * (size_t)blockDim.x + threadIdx.x) >> 5);
    int lane = threadIdx.x & 31;
    if (n >= N) return;
    int head = lane >> 3;
    int cb   = lane * 4;
    float rd = 1.f / (denom[(size_t)n * HEADS + head] + 1e-16f);
    int j0 = rowstart[n], j1 = rowstart[n + 1];
    float4 acc = make_float4(0.f, 0.f, 0.f, 0.f);
    for (int j = j0; j < j1; ++j) {
        int s = esrc[j];                                   // uniform across wave
        float a = esorted[(size_t)j * HEADS + head] * rd;  // coalesced (dst-sorted)
        float4 hv = *reinterpret_cast<const float4*>(H + (size_t)s * OC + cb);
        acc.x += a * hv.x;
        acc.y += a * hv.y;
        acc.z += a * hv.z;
        acc.w += a * hv.w;
    }
    *reinterpret_cast<float4*>(agg + (size_t)n * OC + cb) = acc;
}

// ---------------------------------------------------------------------------
// BN stats / fold / apply
// ---------------------------------------------------------------------------
__global__ __launch_bounds__(128) void bn_stats_kernel(
    const float* __restrict__ agg, const float* __restrict__ bias,
    float* __restrict__ stats, int N) {
    int c = threadIdx.x;
    float b = bias[c];
    float sum = 0.f, sq = 0.f;
    int r0 = blockIdx.x * 128;
    int r1 = min(r0 + 128, N);
    for (int r = r0; r < r1; ++r) {
        float v = agg[(size_t)r * OC + c] + b;
        sum += v;
        sq  += v * v;
    }
    atomicAdd(&stats[c], sum);
    atomicAdd(&stats[OC + c], sq);
}

__global__ void bn_finalize_kernel(const float* __restrict__ stats,
                                   const float* __restrict__ gamma,
                                   const float* __restrict__ beta,
                                   float* __restrict__ ss, int N) {
    int c = threadIdx.x;
    if (c >= OC) return;
    float mean = stats[c] / (float)N;
    float var  = stats[OC + c] / (float)N - mean * mean;
    float inv  = rsqrtf(var + EPS_BN);
    float sc   = inv * gamma[c];
    ss[c]      = sc;
    ss[OC + c] = beta[c] - mean * sc;
}

__global__ void bn_apply_kernel(const float* __restrict__ agg,
                                const float* __restrict__ bias,
                                const float* __restrict__ ss,
                                const float* __restrict__ skip,
                                float* __restrict__ out, int total) {
    int idx = blockIdx.x * blockDim.x + threadIdx.x;
    if (idx >= total) return;
    int c = idx & (OC - 1);
    float v = (agg[idx] + bias[c]) * ss[c] + ss[OC + c];
    if (skip) v += skip[idx];
    out[idx] = fmaxf(v, 0.f);
}

// ---------------------------------------------------------------------------
extern "C" void kernel_launch(void* const* d_in, const int* in_sizes, int n_in,
                              void* d_out, int out_size, void* d_ws, size_t ws_size,
                              hipStream_t stream) {
    const float* x   = (const float*)d_in[0];
    const int*   ei  = (const int*)d_in[1];
    const float* W1  = (const float*)d_in[2];
    const float* as1 = (const float*)d_in[3];
    const float* ad1 = (const float*)d_in[4];
    const float* b1  = (const float*)d_in[5];
    const float* g1  = (const float*)d_in[6];
    const float* be1 = (const float*)d_in[7];
    const float* W2  = (const float*)d_in[8];
    const float* as2 = (const float*)d_in[9];
    const float* ad2 = (const float*)d_in[10];
    const float* b2  = (const float*)d_in[11];
    const float* g2  = (const float*)d_in[12];
    const float* be2 = (const float*)d_in[13];

    const int N  = in_sizes[0] / OC;
    const int E  = in_sizes[1] / 2;
    const int EN = E + N;
    const int* src = ei;
    const int* dst = ei + E;

    char* ws = (char*)d_ws;
    size_t off = 0;
    auto alloc = [&](size_t bytes) -> char* {
        char* p = ws + off;
        off = (off + bytes + 255) & ~(size_t)255;
        return p;
    };
    float*    h        = (float*)alloc((size_t)N * OC * 4);
    float*    agg      = (float*)alloc((size_t)N * OC * 4);
    float*    as_      = (float*)alloc((size_t)N * HEADS * 4);
    float*    ad_      = (float*)alloc((size_t)N * HEADS * 4);
    unsigned* mmax     = (unsigned*)alloc((size_t)N * HEADS * 4);
    float*    denom    = (float*)alloc((size_t)N * HEADS * 4);
    float*    esorted  = (float*)alloc((size_t)EN * HEADS * 4);
    int*      esrc     = (int*)alloc((size_t)EN * 4);
    int*      deg      = (int*)alloc((size_t)N * 4);
    int*      rowstart = (int*)alloc((size_t)(N + 1) * 4);
    int*      cursor   = (int*)alloc((size_t)N * 4);
    const int nb = (N + 255) / 256;
    int*      partial  = (int*)alloc((size_t)nb * 4);
    int*      partOff  = (int*)alloc((size_t)nb * 4);
    float*    stats    = (float*)alloc(2 * OC * 4);
    float*    ss       = (float*)alloc(2 * OC * 4);
    float*    y1       = (float*)d_out;   // layer-1 activations staged in d_out

    const int gGemm  = (N + 63) / 64;
    const int gAlpha = (int)(((size_t)N * 32 + 255) / 256);
    const int gEdge  = (EN + 255) / 256;
    const int gAggW  = (int)(((size_t)N * 32 + 255) / 256);
    const int gStats = (N + 127) / 128;
    const int gElem  = (int)(((size_t)N * OC + 255) / 256);

    // ---- CSR build (topology depends only on dst; shared by both layers) ----
    hipMemsetAsync(deg, 0, (size_t)N * 4, stream);
    deg_count_kernel<<<gEdge, 256, 0, stream>>>(dst, deg, E, EN);
    scan_partial_kernel<<<nb, 256, 0, stream>>>(deg, partial, N);
    scan_offsets_kernel<<<1, 512, 0, stream>>>(partial, partOff, nb, rowstart, N, EN);
    scan_write_kernel<<<nb, 256, 0, stream>>>(deg, partOff, rowstart, N);

    for (int layer = 0; layer < 2; ++layer) {
        const float* Xin   = (layer == 0) ? x   : y1;
        const float* W     = (layer == 0) ? W1  : W2;
        const float* asv   = (layer == 0) ? as1 : as2;
        const float* adv   = (layer == 0) ? ad1 : ad2;
        const float* bias  = (layer == 0) ? b1  : b2;
        const float* gamma = (layer == 0) ? g1  : g2;
        const float* beta  = (layer == 0) ? be1 : be2;
        const float* skip  = (layer == 0) ? nullptr : x;

        hipMemsetAsync(mmax,  0, (size_t)N * HEADS * 4, stream);
        hipMemsetAsync(denom, 0, (size_t)N * HEADS * 4, stream);
        hipMemsetAsync(stats, 0, 2 * OC * 4, stream);
        hipMemcpyAsync(cursor, rowstart, (size_t)N * 4, hipMemcpyDeviceToDevice, stream);

        gemm_wmma_kernel<<<gGemm, 256, 0, stream>>>(Xin, W, h, N);
        alphas_kernel<<<gAlpha, 256, 0, stream>>>(h, asv, adv, as_, ad_, N);
        edge_max_kernel<<<gEdge, 256, 0, stream>>>(src, dst, as_, ad_, mmax, E, EN);
        edge_exp_fill_kernel<<<gEdge, 256, 0, stream>>>(src, dst, as_, ad_, mmax, denom,
                                                        cursor, esorted, esrc, E, EN);
        agg_csr_kernel<<<gAggW, 256, 0, stream>>>(rowstart, esrc, esorted, denom, h, agg, N);
        bn_stats_kernel<<<gStats, 128, 0, stream>>>(agg, bias, stats, N);
        bn_finalize_kernel<<<1, 128, 0, stream>>>(stats, gamma, beta, ss, N);
        bn_apply_kernel<<<gElem, 256, 0, stream>>>(agg, bias, ss, skip, y1, N * OC);
    }
}